// DecoderLSTMWithAttn_3143916061082
// MI455X (gfx1250) — compile-verified
//
#include <hip/hip_runtime.h>
#include <hip/hip_bf16.h>
#include <math.h>

typedef __attribute__((ext_vector_type(2))) float v2f;
typedef __attribute__((ext_vector_type(8))) float v8f;

#define NB 64      // batch
#define HH 1024
#define EE 512
#define LL 128
#define VV 32000

__device__ __forceinline__ v8f wmma_f32(v2f a, v2f b, v8f c) {
    return __builtin_amdgcn_wmma_f32_16x16x4_f32(false, a, false, b, (short)0, c,
                                                 false, false);
}

// --- CDNA5 async global->LDS copy (ASYNCcnt tracked), with safe fallback ----
#if __has_builtin(__builtin_amdgcn_global_load_async_to_lds_b128)
#define HAS_ASYNC_LDS 1
#else
#define HAS_ASYNC_LDS 0
#endif

typedef int v4i_ __attribute__((vector_size(16)));
typedef __attribute__((address_space(1))) v4i_* gptr4;  // global (AS1)
typedef __attribute__((address_space(3))) v4i_* lptr4;  // LDS (AS3)

__device__ __forceinline__ void stage16(const float* __restrict__ g,
                                        float* __restrict__ l) {
#if HAS_ASYNC_LDS
    __builtin_amdgcn_global_load_async_to_lds_b128((gptr4)g, (lptr4)l, 0, 0);
#else
    *(float4*)l = *(const float4*)g;
#endif
}

__device__ __forceinline__ void stage_join() {
#if HAS_ASYNC_LDS && __has_builtin(__builtin_amdgcn_s_wait_asynccnt)
    __builtin_amdgcn_s_wait_asynccnt(0);
#endif
}

// ---------------------------------------------------------------------------
// Kernel 1: embedding gather + pack h_prev into gates-activation buffer
// Xg[n] = [ y_t(512) | ctx(1024, written later) | h_prev(1024) ]
// ---------------------------------------------------------------------------
__global__ __launch_bounds__(256) void gather_pack(
    const int* __restrict__ ids, const float* __restrict__ embed_W,
    const float* __restrict__ h_prev, float* __restrict__ Xg) {
    const int n = blockIdx.x;
    const int tid = threadIdx.x;
    const int id = ids[n];
    const float* er = embed_W + (size_t)id * EE;
    float* xr = Xg + (size_t)n * 2560;
    for (int e = tid; e < EE; e += 256) xr[e] = er[e];
    for (int h = tid; h < HH; h += 256) xr[1536 + h] = h_prev[(size_t)n * HH + h];
}

// ---------------------------------------------------------------------------
// Generic tall-skinny WMMA GEMM: out[64, N] = X[64, Ktot] @ [W0 | W1]^T + bias
// W0: [N, K0] row-major, W1: [N, K1] row-major (K1 may be 0).
// Block = 128 threads (4 waves); each wave computes a 64x16 output strip.
// X panels (64x64) staged (async) in LDS, shared by all 4 waves; W streamed.
// ---------------------------------------------------------------------------
__global__ __launch_bounds__(128) void strip_gemm(
    const float* __restrict__ X, int Ktot,
    const float* __restrict__ W0, int K0,
    const float* __restrict__ W1, int K1,
    const float* __restrict__ bias,
    float* __restrict__ out, int N) {
    __shared__ float Xs[64 * 68];
    const int tid  = threadIdx.x;
    const int wave = tid >> 5;
    const int lane = tid & 31;
    const int l16  = lane & 15;
    const int koff = (lane >> 4) * 2;            // lanes 16..31 take K+2,K+3
    const int n0   = blockIdx.x * 64 + wave * 16;
    const int wrow = n0 + l16;

    v8f c0 = {}, c1 = {}, c2 = {}, c3 = {};

    for (int seg = 0; seg < 2; ++seg) {
        const int Kseg = seg ? K1 : K0;
        if (Kseg == 0) continue;
        const float* W = seg ? W1 : W0;
        const int xbase = seg ? K0 : 0;
        const float* wr = W + (size_t)wrow * Kseg;
        for (int kp = 0; kp < Kseg; kp += 64) {
            __syncthreads();
            // stage X[:, xbase+kp .. +63] -> LDS (64 rows x 64 cols)
            for (int i = tid; i < 64 * 16; i += 128) {
                const int r  = i >> 4;
                const int cs = (i & 15) * 4;
                stage16(X + (size_t)r * Ktot + xbase + kp + cs, &Xs[r * 68 + cs]);
            }
            stage_join();
            __syncthreads();
            // prefetch next W panel for this lane's weight row
            if (kp + 64 < Kseg) __builtin_prefetch(wr + kp + 64, 0, 0);
#pragma unroll
            for (int kk = 0; kk < 64; kk += 16) {
                v2f b[4], a0[4], a1[4], a2[4], a3[4];
#pragma unroll
                for (int u = 0; u < 4; ++u) {
                    const int kc = kk + 4 * u + koff;
                    b[u]  = *(const v2f*)(wr + kp + kc);
                    a0[u] = *(const v2f*)(&Xs[(l16)      * 68 + kc]);
                    a1[u] = *(const v2f*)(&Xs[(16 + l16) * 68 + kc]);
                    a2[u] = *(const v2f*)(&Xs[(32 + l16) * 68 + kc]);
                    a3[u] = *(const v2f*)(&Xs[(48 + l16) * 68 + kc]);
                }
#pragma unroll
                for (int u = 0; u < 4; ++u) {
                    c0 = wmma_f32(a0[u], b[u], c0);
                    c1 = wmma_f32(a1[u], b[u], c1);
                    c2 = wmma_f32(a2[u], b[u], c2);
                    c3 = wmma_f32(a3[u], b[u], c3);
                }
            }
        }
    }

    const float bv  = bias ? bias[wrow] : 0.f;
    const int   col = wrow;
    const int   rb  = (lane >> 4) * 8;
#pragma unroll
    for (int r = 0; r < 8; ++r) {
        out[(size_t)(0  + rb + r) * N + col] = c0[r] + bv;
        out[(size_t)(16 + rb + r) * N + col] = c1[r] + bv;
        out[(size_t)(32 + rb + r) * N + col] = c2[r] + bv;
        out[(size_t)(48 + rb + r) * N + col] = c3[r] + bv;
    }
}

// ---------------------------------------------------------------------------
// Fused attention scores: scores[n,l] = sum_h v[h]*tanh( (enc@Wk^T)[n,l,h] + q[n,h] )
// Grid: 128 blocks (2 per n: 64 (n,l) rows each). Block = 256 threads (8 waves).
// Wave computes 64 rows x 16 h-cols per strip iteration (8 iterations -> all H),
// enc panel (64x128) staged (async) in LDS, Wk streamed (read once per block).
// ---------------------------------------------------------------------------
__global__ __launch_bounds__(256) void attn_scores(
    const float* __restrict__ enc, const float* __restrict__ Wk,
    const float* __restrict__ q, const float* __restrict__ vw,
    const unsigned char* __restrict__ mask, float* __restrict__ scores) {
    __shared__ float As[64 * 132];
    __shared__ float sAcc[64];
    const int tid  = threadIdx.x;
    const int wave = tid >> 5;
    const int lane = tid & 31;
    const int l16  = lane & 15;
    const int koff = (lane >> 4) * 2;
    const int n    = blockIdx.x >> 1;
    const int l0   = (blockIdx.x & 1) * 64;
    const float* encR = enc + ((size_t)n * LL + l0) * HH;

    if (tid < 64) sAcc[tid] = 0.f;

    float acc[4][8];
#pragma unroll
    for (int mt = 0; mt < 4; ++mt)
#pragma unroll
        for (int r = 0; r < 8; ++r) acc[mt][r] = 0.f;

    for (int s = 0; s < 8; ++s) {
        const int h0 = (s * 8 + wave) * 16;
        const float* wr = Wk + (size_t)(h0 + l16) * HH;
        v8f c0 = {}, c1 = {}, c2 = {}, c3 = {};
        for (int kp = 0; kp < HH; kp += 128) {
            __syncthreads();
            for (int i = tid; i < 64 * 32; i += 256) {
                const int r  = i >> 5;
                const int cs = (i & 31) * 4;
                stage16(encR + (size_t)r * HH + kp + cs, &As[r * 132 + cs]);
            }
            stage_join();
            __syncthreads();
            if (kp + 128 < HH) __builtin_prefetch(wr + kp + 128, 0, 0);
#pragma unroll
            for (int kk = 0; kk < 128; kk += 16) {
                v2f b[4], a0[4], a1[4], a2[4], a3[4];
#pragma unroll
                for (int u = 0; u < 4; ++u) {
                    const int kc = kk + 4 * u + koff;
                    b[u]  = *(const v2f*)(wr + kp + kc);
                    a0[u] = *(const v2f*)(&As[(l16)      * 132 + kc]);
                    a1[u] = *(const v2f*)(&As[(16 + l16) * 132 + kc]);
                    a2[u] = *(const v2f*)(&As[(32 + l16) * 132 + kc]);
                    a3[u] = *(const v2f*)(&As[(48 + l16) * 132 + kc]);
                }
#pragma unroll
                for (int u = 0; u < 4; ++u) {
                    c0 = wmma_f32(a0[u], b[u], c0);
                    c1 = wmma_f32(a1[u], b[u], c1);
                    c2 = wmma_f32(a2[u], b[u], c2);
                    c3 = wmma_f32(a3[u], b[u], c3);
                }
            }
        }
        // epilogue: e = tanh(k + q)*v, accumulate over this strip's 16 h-cols
        const float qv = q[(size_t)n * HH + h0 + l16];
        const float vv = vw[h0 + l16];
#pragma unroll
        for (int r = 0; r < 8; ++r) {
            acc[0][r] += tanhf(c0[r] + qv) * vv;
            acc[1][r] += tanhf(c1[r] + qv) * vv;
            acc[2][r] += tanhf(c2[r] + qv) * vv;
            acc[3][r] += tanhf(c3[r] + qv) * vv;
        }
    }
    // butterfly reduce over the 16 h-columns held by lanes [0..15] / [16..31]
#pragma unroll
    for (int mt = 0; mt < 4; ++mt)
#pragma unroll
        for (int r = 0; r < 8; ++r) {
            float v = acc[mt][r];
            v += __shfl_xor(v, 1, 32);
            v += __shfl_xor(v, 2, 32);
            v += __shfl_xor(v, 4, 32);
            v += __shfl_xor(v, 8, 32);
            acc[mt][r] = v;
        }
    if (l16 == 0) {  // lanes 0 (M=r) and 16 (M=r+8) of each wave
        const int rb = (lane >> 4) * 8;
#pragma unroll
        for (int mt = 0; mt < 4; ++mt)
#pragma unroll
            for (int r = 0; r < 8; ++r)
                atomicAdd(&sAcc[mt * 16 + rb + r], acc[mt][r]);
    }
    __syncthreads();
    if (tid < 64) {
        const int l = l0 + tid;
        float sc = sAcc[tid];
        if (!mask[n * LL + l]) sc = -INFINITY;
        scores[n * LL + l] = sc;
    }
}

// ---------------------------------------------------------------------------
// Softmax over L + context GEMV; writes ctx into both concat activation bufs.
// ---------------------------------------------------------------------------
__global__ __launch_bounds__(256) void softmax_ctx(
    const float* __restrict__ scores, const float* __restrict__ enc,
    float* __restrict__ Xg, float* __restrict__ Xl) {
    __shared__ float s[LL];
    __shared__ float w[LL];
    const int n = blockIdx.x;
    const int tid = threadIdx.x;
    if (tid < LL) s[tid] = scores[n * LL + tid];
    __syncthreads();
    float m = -INFINITY;
    for (int l = 0; l < LL; ++l) m = fmaxf(m, s[l]);
    float sum = 0.f;
    for (int l = 0; l < LL; ++l) sum += expf(s[l] - m);
    if (tid < LL) w[tid] = expf(s[tid] - m) / sum;
    __syncthreads();
    for (int h = tid; h < HH; h += 256) {
        float acc = 0.f;
        for (int l = 0; l < LL; ++l)
            acc += w[l] * enc[((size_t)n * LL + l) * HH + h];
        Xg[(size_t)n * 2560 + EE + h] = acc;        // gates input [y|ctx|h]
        Xl[(size_t)n * 2048 + HH + h] = acc;        // logits input [h_t|ctx]
    }
}

// ---------------------------------------------------------------------------
// LSTM elementwise: gates z[64,4096] -> c_t, h_t; h_t also packed for logits
// ---------------------------------------------------------------------------
__global__ __launch_bounds__(256) void lstm_elem(
    const float* __restrict__ z, const float* __restrict__ c_prev,
    float* __restrict__ out_h, float* __restrict__ out_c,
    float* __restrict__ Xl) {
    const int n = blockIdx.x;
    for (int h = threadIdx.x; h < HH; h += 256) {
        const size_t zb = (size_t)n * 4096;
        const float zi = z[zb + h];
        const float zf = z[zb + 1024 + h];
        const float zg = z[zb + 2048 + h];
        const float zo = z[zb + 3072 + h];
        const float it = 1.f / (1.f + expf(-zi));
        const float ft = 1.f / (1.f + expf(-zf));
        const float gt = tanhf(zg);
        const float ot = 1.f / (1.f + expf(-zo));
        const float ct = ft * c_prev[(size_t)n * HH + h] + it * gt;
        const float ht = ot * tanhf(ct);
        out_c[(size_t)n * HH + h] = ct;
        out_h[(size_t)n * HH + h] = ht;
        Xl[(size_t)n * 2048 + h] = ht;
    }
}

// ---------------------------------------------------------------------------
// In-place log-softmax over V=32000 per row
// ---------------------------------------------------------------------------
__global__ __launch_bounds__(256) void log_softmax_rows(float* __restrict__ logits) {
    __shared__ float red[256];
    const int n = blockIdx.x;
    const int tid = threadIdx.x;
    float* row = logits + (size_t)n * VV;
    float m = -INFINITY;
    for (int v = tid; v < VV; v += 256) m = fmaxf(m, row[v]);
    red[tid] = m;
    __syncthreads();
    for (int s = 128; s > 0; s >>= 1) {
        if (tid < s) red[tid] = fmaxf(red[tid], red[tid + s]);
        __syncthreads();
    }
    m = red[0];
    __syncthreads();
    float sum = 0.f;
    for (int v = tid; v < VV; v += 256) sum += expf(row[v] - m);
    red[tid] = sum;
    __syncthreads();
    for (int s = 128; s > 0; s >>= 1) {
        if (tid < s) red[tid] += red[tid + s];
        __syncthreads();
    }
    const float lse = logf(red[0]) + m;
    __syncthreads();
    for (int v = tid; v < VV; v += 256) row[v] -= lse;
}

// ---------------------------------------------------------------------------
extern "C" void kernel_launch(void* const* d_in, const int* in_sizes, int n_in,
                              void* d_out, int out_size, void* d_ws, size_t ws_size,
                              hipStream_t stream) {
    const int*   y_ids     = (const int*)d_in[0];
    const float* h_prev    = (const float*)d_in[1];
    const float* c_prev    = (const float*)d_in[2];
    const float* enc       = (const float*)d_in[3];
    const unsigned char* mask = (const unsigned char*)d_in[4];
    const float* embed_W   = (const float*)d_in[5];
    const float* Wq        = (const float*)d_in[6];
    const float* Wk        = (const float*)d_in[7];
    const float* v_w       = (const float*)d_in[8];
    const float* x2gates_W = (const float*)d_in[9];
    const float* x2gates_b = (const float*)d_in[10];
    const float* h2gates_W = (const float*)d_in[11];
    const float* h2o_W     = (const float*)d_in[12];
    const float* h2o_b     = (const float*)d_in[13];

    // workspace layout (floats)
    float* ws     = (float*)d_ws;
    float* Xg     = ws;                  // [64][2560] = [y|ctx|h_prev]
    float* Xl     = Xg + NB * 2560;      // [64][2048] = [h_t|ctx]
    float* q      = Xl + NB * 2048;      // [64][1024]
    float* scores = q  + NB * HH;        // [64][128]
    float* z      = scores + NB * LL;    // [64][4096]

    float* out_lp = (float*)d_out;               // [64][32000]
    float* out_h  = out_lp + (size_t)NB * VV;    // [64][1024]
    float* out_c  = out_h + (size_t)NB * HH;     // [64][1024]

    gather_pack<<<NB, 256, 0, stream>>>(y_ids, embed_W, h_prev, Xg);

    // q = h_prev @ Wq^T   [64,1024]
    strip_gemm<<<HH / 64, 128, 0, stream>>>(h_prev, HH, Wq, HH, nullptr, 0,
                                            nullptr, q, HH);

    // attention energies -> masked scores
    attn_scores<<<NB * 2, 256, 0, stream>>>(enc, Wk, q, v_w, mask, scores);

    // softmax + context
    softmax_ctx<<<NB, 256, 0, stream>>>(scores, enc, Xg, Xl);

    // gates: z = [y|ctx] @ x2gates_W^T + x2gates_b + h_prev @ h2gates_W^T
    strip_gemm<<<4096 / 64, 128, 0, stream>>>(Xg, 2560, x2gates_W, 1536,
                                              h2gates_W, HH, x2gates_b, z, 4096);

    lstm_elem<<<NB, 256, 0, stream>>>(z, c_prev, out_h, out_c, Xl);

    // logits straight into d_out: [h_t|ctx] @ h2o_W^T + h2o_b
    strip_gemm<<<VV / 64, 128, 0, stream>>>(Xl, 2048, h2o_W, 2048, nullptr, 0,
                                            h2o_b, out_lp, VV);

    log_softmax_rows<<<NB, 256, 0, stream>>>(out_lp);
}